// Boundary_smoothing_29025388986507
// MI455X (gfx1250) — compile-verified
//
#include <hip/hip_runtime.h>
#include <hip/hip_bf16.h>
#include <math.h>
#include <stdint.h>

#define AS1 __attribute__((address_space(1)))
#define AS3 __attribute__((address_space(3)))

#if __has_builtin(__builtin_amdgcn_global_load_async_to_lds_b128) && \
    __has_builtin(__builtin_amdgcn_s_wait_asynccnt)
#define SB_HAVE_ASYNC 1
#else
#define SB_HAVE_ASYNC 0
#endif

#if __has_builtin(__builtin_amdgcn_wmma_f32_16x16x4_f32)
#define SB_HAVE_WMMA 1
#else
#define SB_HAVE_WMMA 0
#endif

typedef __attribute__((ext_vector_type(2))) float v2f;
typedef __attribute__((ext_vector_type(8))) float v8f;
typedef __attribute__((ext_vector_type(4))) float f4;
typedef __attribute__((ext_vector_type(4))) int i4;
// Pointee type expected by the async-to-LDS builtin (from clang diagnostic).
typedef int v4i __attribute__((vector_size(4 * sizeof(int))));

constexpr int kB = 16, kS = 256, kL = 24;
constexpr int kRow = kS * kL;   // 6144 contiguous elements per (b, i) row
constexpr int kRows = kB * kS;  // 4096 rows -> 4096 blocks
constexpr int kThreads = 256;
constexpr int kPad = kL;        // halo so j-neighbor taps need no branches
constexpr float kEpsPos = 0.1f;   // SB_EPSILON
constexpr float kEpsN = 0.025f;   // SB_EPSILON / (SB_SIZE * dist * 4)
constexpr float kLog2e = 1.4426950408889634f;
constexpr float kLn2 = 0.6931471805599453f;

// softplus(-|x|) = log(1 + exp(-|x|)) on the native transcendental pipe:
// 2 trans + 3 VALU instead of ~50 VALU of inlined ocml expf/log1pf.
__device__ __forceinline__ float softplus_neg(float ax) {
  const float z = __builtin_amdgcn_exp2f(-kLog2e * ax);  // exp(-ax), ax >= 0
  return kLn2 * __builtin_amdgcn_logf(1.0f + z);         // ln(1+z)
}

// Full wave32 sum, result replicated to every lane.
// Matrix pipe: A(16x4) holds the 32 lane partials (K0/K2 slots), B = ones,
// D row-sums pair lanes (m, m+16); 7 in-lane adds + 1 shfl finish.
__device__ __forceinline__ float wave_sum32(float v) {
#if SB_HAVE_WMMA
  v2f a;
  a.x = v;
  a.y = 0.0f;
  v2f ones;
  ones.x = 1.0f;
  ones.y = 1.0f;
  v8f c = {};
  c = __builtin_amdgcn_wmma_f32_16x16x4_f32(false, a, false, ones, (short)0, c,
                                            false, false);
  float s = ((c[0] + c[1]) + (c[2] + c[3])) + ((c[4] + c[5]) + (c[6] + c[7]));
  return s + __shfl_xor(s, 16, 32);
#else
  for (int m = 16; m > 0; m >>= 1) v += __shfl_xor(v, m, 32);
  return v;
#endif
}

__global__ __launch_bounds__(kThreads) void sb_bce_row_kernel(
    const float* __restrict__ predict, const float* __restrict__ target,
    const int* __restrict__ mask, float* __restrict__ part_bce,
    float* __restrict__ part_msk) {
  __shared__ float sT[kRow + 2 * kPad];  // zero halo both sides
  __shared__ int sM[kRow + 2 * kPad];
  __shared__ float redB[kThreads / 32];
  __shared__ float redM[kThreads / 32];

  const int bid = blockIdx.x;    // bid = b*kS + i
  const int i = bid & (kS - 1);  // row index within the S x S plane
  const int tid = threadIdx.x;
  const size_t base = (size_t)bid * kRow;

  const float* pr = predict + base;
  const float* t0 = target + base;
  const int* m0 = mask + base;
  const bool hasPrev = (i > 0);
  const bool hasNext = (i < kS - 1);
  // Clamp neighbor-row pointers so loads are always in-bounds; the 0/1
  // factors kill out-of-range contributions (zero-padded shift semantics).
  const float* tmL = hasPrev ? (t0 - kRow) : t0;
  const float* tpL = hasNext ? (t0 + kRow) : t0;
  const int* mmL = hasPrev ? (m0 - kRow) : m0;
  const int* mpL = hasNext ? (m0 + kRow) : m0;
  const float fPrev = hasPrev ? 1.f : 0.f;
  const float fNext = hasNext ? 1.f : 0.f;

  // Zero the halos (24 floats = 6 b128 chunks per side).
  if (tid < kPad / 4) {
    const f4 zf = 0.0f;
    const i4 zi = 0;
    *(f4*)(sT + 4 * tid) = zf;
    *(f4*)(sT + kPad + kRow + 4 * tid) = zf;
    *(i4*)(sM + 4 * tid) = zi;
    *(i4*)(sM + kPad + kRow + 4 * tid) = zi;
  }

  // ---- Stage center target/mask row into LDS (async DMA path) ----
#if SB_HAVE_ASYNC
  for (int v = tid; v < kRow / 4; v += kThreads) {
    __builtin_amdgcn_global_load_async_to_lds_b128(
        (AS1 v4i*)(uintptr_t)(t0 + 4 * v),
        (AS3 v4i*)(uint32_t)(uintptr_t)(&sT[kPad + 4 * v]), 0, 0);
    __builtin_amdgcn_global_load_async_to_lds_b128(
        (AS1 v4i*)(uintptr_t)(m0 + 4 * v),
        (AS3 v4i*)(uint32_t)(uintptr_t)(&sM[kPad + 4 * v]), 0, 0);
  }
  __builtin_amdgcn_s_wait_asynccnt(0);
#else
  for (int v = tid; v < kRow / 4; v += kThreads) {
    *(f4*)(sT + kPad + 4 * v) = *(const f4*)(t0 + 4 * v);
    *(i4*)(sM + kPad + 4 * v) = *(const i4*)(m0 + 4 * v);
  }
#endif
  __syncthreads();

  const float* sTc = sT + kPad;
  const int* sMc = sM + kPad;

  float accB = 0.f, accM = 0.f;
#pragma unroll
  for (int u = 0; u < kRow / 4 / kThreads; ++u) {  // 6 iterations of float4
    const int q = tid + u * kThreads;  // float4 index in the row
    const int e = 4 * q;               // element index; e % 4 == 0
    // predict is streamed exactly once: non-temporal so it does not evict
    // target/mask lines that neighbor blocks re-hit in L2.
    const f4 x4 = __builtin_nontemporal_load((const f4*)pr + q);
    // All four elements of a float4 share the same j (kL % 4 == 0), and the
    // zeroed halo makes the +/-kL taps unconditional b128 LDS loads.
    const f4 tC4 = *(const f4*)(sTc + e);
    const f4 tJm4 = *(const f4*)(sTc + e - kL);
    const f4 tJp4 = *(const f4*)(sTc + e + kL);
    const i4 mC4 = *(const i4*)(sMc + e);
    const i4 mJm4 = *(const i4*)(sMc + e - kL);
    const i4 mJp4 = *(const i4*)(sMc + e + kL);
    const f4 tIm4 = *(const f4*)(tmL + e);
    const f4 tIp4 = *(const f4*)(tpL + e);
    const i4 mIm4 = *(const i4*)(mmL + e);
    const i4 mIp4 = *(const i4*)(mpL + e);

#pragma unroll
    for (int c = 0; c < 4; ++c) {
      // target is exactly {0,1} (reference tests target==1), mask is {0,1}.
      const float Mf = (float)mC4[c];
      const float P = tC4[c];
      const float add =
          tJm4[c] + tJp4[c] + fPrev * tIm4[c] + fNext * tIp4[c];
      const float cnt = (float)mJm4[c] + (float)mJp4[c] +
                        fPrev * (float)mIm4[c] + fNext * (float)mIp4[c];
      // b2l = target - eps*P + epsN*(add*M + P*(4 - cnt)); target == P here.
      const float b2l =
          (1.f - kEpsPos) * P + kEpsN * (add * Mf + P * (4.f - cnt));
      const float x = x4[c];
      // numerically-stable BCE with logits
      const float bce = fmaxf(x, 0.f) - x * b2l + softplus_neg(fabsf(x));
      accB += bce * Mf;
      accM += Mf;
    }
  }

  // Wave reduction on the matrix pipe (EXEC is all-ones here: no divergence).
  const float wB = wave_sum32(accB);
  const float wM = wave_sum32(accM);
  const int wave = tid >> 5;
  if ((tid & 31) == 0) {
    redB[wave] = wB;
    redM[wave] = wM;
  }
  __syncthreads();
  if (tid == 0) {
    float sb = 0.f, sm = 0.f;
#pragma unroll
    for (int w = 0; w < kThreads / 32; ++w) {
      sb += redB[w];
      sm += redM[w];
    }
    part_bce[bid] = sb;  // every block writes its slot: d_ws fully initialized
    part_msk[bid] = sm;
  }
}

__global__ __launch_bounds__(kThreads) void sb_finalize_kernel(
    const float* __restrict__ part_bce, const float* __restrict__ part_msk,
    float* __restrict__ out) {
  __shared__ float rb[kThreads];
  __shared__ float rm[kThreads];
  float sb = 0.f, sm = 0.f;
  for (int v = threadIdx.x; v < kRows; v += kThreads) {
    sb += part_bce[v];
    sm += part_msk[v];
  }
  rb[threadIdx.x] = sb;
  rm[threadIdx.x] = sm;
  __syncthreads();
  for (int off = kThreads / 2; off > 0; off >>= 1) {
    if ((int)threadIdx.x < off) {
      rb[threadIdx.x] += rb[threadIdx.x + off];
      rm[threadIdx.x] += rm[threadIdx.x + off];
    }
    __syncthreads();
  }
  if (threadIdx.x == 0) out[0] = rb[0] / rm[0];
}

extern "C" void kernel_launch(void* const* d_in, const int* in_sizes, int n_in,
                              void* d_out, int out_size, void* d_ws,
                              size_t ws_size, hipStream_t stream) {
  (void)in_sizes;
  (void)n_in;
  (void)out_size;
  (void)ws_size;  // needs only 2*4096 floats = 32 KB of scratch
  const float* predict = (const float*)d_in[0];
  const float* target = (const float*)d_in[1];
  const int* mask = (const int*)d_in[2];
  float* part_bce = (float*)d_ws;
  float* part_msk = part_bce + kRows;

  sb_bce_row_kernel<<<kRows, kThreads, 0, stream>>>(predict, target, mask,
                                                    part_bce, part_msk);
  sb_finalize_kernel<<<1, kThreads, 0, stream>>>(part_bce, part_msk,
                                                 (float*)d_out);
}